// GCNLinkPredictor_22935125361178
// MI455X (gfx1250) — compile-verified
//
#include <hip/hip_runtime.h>

#define NN 50000
#define NE 800000
#define DD 128

typedef float v2f __attribute__((ext_vector_type(2)));
typedef float v8f __attribute__((ext_vector_type(8)));

// ---------------- degree / normalization ----------------

__global__ void k_fill(float* __restrict__ p, float v, int n) {
    int i = blockIdx.x * blockDim.x + threadIdx.x;
    if (i < n) p[i] = v;
}

__global__ void k_count_deg(const long long* __restrict__ ei, float* __restrict__ deg) {
    int e = blockIdx.x * blockDim.x + threadIdx.x;
    if (e < NE) {
        int t = (int)ei[NE + e];          // row 1 = dst
        atomicAdd(&deg[t], 1.0f);
    }
}

__global__ void k_rsqrt_inplace(float* __restrict__ p, int n) {
    int i = blockIdx.x * blockDim.x + threadIdx.x;
    if (i < n) p[i] = rsqrtf(p[i]);
}

// ---------------- fp32 WMMA GEMM: H[M,128] = A[M,128] @ W[128,128] ----------------
// Block = 256 threads (8 waves). Block handles a 16-row strip; wave w owns
// columns [16w, 16w+16). K-loop: 32 x V_WMMA_F32_16X16X4_F32.
//
// Fragment layouts per CDNA5 ISA 7.12.2 (32-bit, wave32):
//   A 16x4 : lane&15 = M ; VGPR{0,1} = K{koff, koff+1}, koff = (lane>>4)*2
//   B 4x16 : lane&15 = N ; VGPR{0,1} = K{koff, koff+1}
//   C/D    : lane&15 = N ; VGPR j -> M = j + (lane>>4)*8

template <bool RELU>
__global__ __launch_bounds__(256) void k_gemm_wmma(const float* __restrict__ A,
                                                   const float* __restrict__ W,
                                                   float* __restrict__ H) {
    const int lane = threadIdx.x & 31;
    const int wave = threadIdx.x >> 5;
    const int half = lane >> 4;        // 0 or 1
    const int l15  = lane & 15;
    const int m0   = blockIdx.x * 16;  // N=50000 = 3125*16, exact
    const int koff = half * 2;
    const int n    = wave * 16 + l15;

    const float* __restrict__ Arow = A + (size_t)(m0 + l15) * DD;

    v8f acc = {};
#pragma unroll
    for (int k = 0; k < DD; k += 4) {
        v2f a = *(const v2f*)(Arow + k + koff);          // 8B aligned (koff even)
        if (RELU) { a.x = fmaxf(a.x, 0.0f); a.y = fmaxf(a.y, 0.0f); }
        v2f b;
        b.x = W[(size_t)(k + koff) * DD + n];
        b.y = W[(size_t)(k + koff + 1) * DD + n];
        acc = __builtin_amdgcn_wmma_f32_16x16x4_f32(
            /*neg_a=*/false, a, /*neg_b=*/false, b,
            /*c_mod=*/(short)0, acc, /*reuse_a=*/false, /*reuse_b=*/false);
    }
#pragma unroll
    for (int j = 0; j < 8; ++j) {
        int m = m0 + half * 8 + j;
        H[(size_t)m * DD + n] = acc[j];
    }
}

// ---------------- self-loop + bias init: out = h * dinv^2 + bias ----------------

__global__ void k_self_bias(const float* __restrict__ h, const float* __restrict__ dinv,
                            const float* __restrict__ bias, float* __restrict__ out) {
    int i = blockIdx.x * blockDim.x + threadIdx.x;   // over N*D = 6.4M
    if (i < NN * DD) {
        int node = i >> 7;
        int d    = i & (DD - 1);
        float di = dinv[node];
        out[i] = h[i] * di * di + bias[d];
    }
}

// ---------------- edge scatter: agg[dst] += dinv[src]*dinv[dst] * h[src] ----------------
// One lane per 4 dims -> 32 lanes cover one edge's 128 features.

__global__ void k_scatter(const long long* __restrict__ ei, const float* __restrict__ h,
                          const float* __restrict__ dinv, float* __restrict__ agg) {
    int tid = blockIdx.x * blockDim.x + threadIdx.x;  // NE*32 = 25.6M, fits int
    if (tid < NE * 32) {
        int e  = tid >> 5;
        int d4 = (tid & 31) << 2;
        int s  = (int)ei[e];
        int t  = (int)ei[NE + e];
        float nm = dinv[s] * dinv[t];
        float4 hv = *(const float4*)(h + (size_t)s * DD + d4);
        float* dst = agg + (size_t)t * DD + d4;
        atomicAdd(dst + 0, hv.x * nm);
        atomicAdd(dst + 1, hv.y * nm);
        atomicAdd(dst + 2, hv.z * nm);
        atomicAdd(dst + 3, hv.w * nm);
    }
}

// ---------------- launcher ----------------

extern "C" void kernel_launch(void* const* d_in, const int* in_sizes, int n_in,
                              void* d_out, int out_size, void* d_ws, size_t ws_size,
                              hipStream_t stream) {
    const float*     x  = (const float*)d_in[0];
    const long long* ei = (const long long*)d_in[1];   // int64 [2, E]
    const float*     W1 = (const float*)d_in[2];
    const float*     b1 = (const float*)d_in[3];
    const float*     W2 = (const float*)d_in[4];
    const float*     b2 = (const float*)d_in[5];
    float* out = (float*)d_out;

    char* ws = (char*)d_ws;
    float* dinv = (float*)ws;                                  // N floats (deg -> dinv)
    size_t off = ((size_t)NN * sizeof(float) + 255) & ~(size_t)255;
    float* h    = (float*)(ws + off);                          // N*128 floats
    float* agg  = h + (size_t)NN * DD;                         // N*128 floats

    const int T = 256;
    // normalization
    k_fill<<<(NN + T - 1) / T, T, 0, stream>>>(dinv, 1.0f, NN);        // self-loop
    k_count_deg<<<(NE + T - 1) / T, T, 0, stream>>>(ei, dinv);
    k_rsqrt_inplace<<<(NN + T - 1) / T, T, 0, stream>>>(dinv, NN);

    // layer 1
    k_gemm_wmma<false><<<NN / 16, T, 0, stream>>>(x, W1, h);
    k_self_bias<<<(NN * DD + T - 1) / T, T, 0, stream>>>(h, dinv, b1, agg);
    k_scatter<<<(NE * 32 + T - 1) / T, T, 0, stream>>>(ei, h, dinv, agg);

    // layer 2 (ReLU fused into GEMM A-load); h buffer reused for h2
    k_gemm_wmma<true><<<NN / 16, T, 0, stream>>>(agg, W2, h);
    k_self_bias<<<(NN * DD + T - 1) / T, T, 0, stream>>>(h, dinv, b2, out);
    k_scatter<<<(NE * 32 + T - 1) / T, T, 0, stream>>>(ei, h, dinv, out);
}